// NLRidge_46866683134061
// MI455X (gfx1250) — compile-verified
//
#include <hip/hip_runtime.h>

// NLRidge on MI455X (gfx1250, wave32).
// Everything fits in L2 (192MB); ~25 GFLOP total -> latency bound, so the
// per-group pipeline (gather -> Q=M M^T via V_WMMA_F32_16X16X4_F32 -> LDS
// Gauss-Jordan inverse -> theta -> theta@Y via WMMA -> atomic aggregation)
// is fused into a single 128-thread workgroup per group. f32 WMMA is chosen
// (not f16/bf16) because Q feeds a 55x55 matrix inversion.

typedef __attribute__((ext_vector_type(2))) float v2f;
typedef __attribute__((ext_vector_type(8))) float v8f;

#define BS 128
#define IMG_W 256
#define IMG_HW 65536
#define WC 250          // H - p + 1
#define PADW 292        // 256 + 2*18
#define NRMW 286        // PADW - 7 + 1
#define NPOS 64
#define NIMG 4

__device__ __forceinline__ v8f wmma4(v2f a, v2f b, v8f c) {
  // D = A(16x4 f32) x B(4x16 f32) + C(16x16 f32)
  return __builtin_amdgcn_wmma_f32_16x16x4_f32(false, a, false, b, (short)0, c,
                                               false, false);
}

__device__ __forceinline__ int posAt(int i) { return i < 63 ? 4 * i : 249; }

// ---------------------------------------------------------------- utilities
__global__ void fill_kernel(float* p, float v, int n) {
  int i = blockIdx.x * blockDim.x + threadIdx.x;
  if (i < n) p[i] = v;
}

__global__ void pad_kernel(const float* __restrict__ src, float* __restrict__ dst) {
  int i = blockIdx.x * blockDim.x + threadIdx.x;
  if (i >= NIMG * PADW * PADW) return;
  int img = i / (PADW * PADW);
  int rc = i % (PADW * PADW);
  int r = rc / PADW, c = rc % PADW;
  float v = 0.f;
  if (r >= 18 && r < 18 + IMG_W && c >= 18 && c < 18 + IMG_W)
    v = src[img * IMG_HW + (r - 18) * IMG_W + (c - 18)];
  dst[i] = v;
}

// norm[r][c] = mean of squares of 7x7 patch of padded image at (r,c)
__global__ void norm_kernel(const float* __restrict__ pad, float* __restrict__ nrm) {
  int i = blockIdx.x * blockDim.x + threadIdx.x;
  if (i >= NIMG * NRMW * NRMW) return;
  int img = i / (NRMW * NRMW);
  int rc = i % (NRMW * NRMW);
  int r = rc / NRMW, c = rc % NRMW;
  const float* P = pad + (size_t)img * PADW * PADW;
  float s = 0.f;
#pragma unroll
  for (int a = 0; a < 7; ++a)
#pragma unroll
    for (int b = 0; b < 7; ++b) {
      float t = P[(r + a) * PADW + (c + b)];
      s += t * t;
    }
  nrm[i] = s * (1.f / 49.f);
}

// ---------------------------------------------------------- block matching
// One workgroup per (ih, iw, img). Window (43x43) cached in LDS; each thread
// evaluates a strided set of the 1369 candidates; then k rounds of argmin.
template <int KSEL>
__global__ __launch_bounds__(BS) void bm_kernel(const float* __restrict__ padimg,
                                                const float* __restrict__ nrm,
                                                int* __restrict__ ind) {
  __shared__ float win[43 * 43];
  __shared__ float dist[37 * 37];
  __shared__ float redV[BS];
  __shared__ int redI[BS];
  const float INF = __builtin_inff();
  int tid = threadIdx.x;
  int ih = posAt(blockIdx.x), iw = posAt(blockIdx.y);
  int img = blockIdx.z;
  const float* P = padimg + (size_t)img * PADW * PADW;
  for (int i = tid; i < 43 * 43; i += BS) {
    int r = i / 43, c = i % 43;
    win[i] = P[(ih + r) * PADW + (iw + c)];
  }
  __syncthreads();
  for (int cand = tid; cand < 37 * 37; cand += BS) {
    int dr = cand / 37, dc = cand % 37;
    float sp = 0.f;
#pragma unroll
    for (int a = 0; a < 7; ++a)
#pragma unroll
      for (int b = 0; b < 7; ++b)
        sp += win[(dr + a) * 43 + (dc + b)] * win[(18 + a) * 43 + (18 + b)];
    float dd = nrm[(size_t)img * NRMW * NRMW + (ih + dr) * NRMW + (iw + dc)] -
               sp * (2.f / 49.f);
    int rr = ih + dr - 18, cc = iw + dc - 18;
    if (rr < 0 || rr > 249 || cc < 0 || cc > 249) dd = INF;
    if (dr == 18 && dc == 18) dd = -INF;  // self patch always first
    dist[cand] = dd;
  }
  __syncthreads();
  int g = (img * NPOS + blockIdx.x) * NPOS + blockIdx.y;
  for (int j = 0; j < KSEL; ++j) {
    float bv = INF;
    int bi = 0;
    for (int cand = tid; cand < 37 * 37; cand += BS)
      if (dist[cand] < bv) { bv = dist[cand]; bi = cand; }
    redV[tid] = bv;
    redI[tid] = bi;
    __syncthreads();
    for (int off = BS / 2; off > 0; off >>= 1) {
      if (tid < off && redV[tid + off] < redV[tid]) {
        redV[tid] = redV[tid + off];
        redI[tid] = redI[tid + off];
      }
      __syncthreads();
    }
    if (tid == 0) {
      int w = redI[0];
      dist[w] = INF;
      int ir = w / 37 - 18 + ih;
      int ic = w % 37 - 18 + iw;
      ind[(size_t)g * KSEL + j] = ir * WC + ic;
    }
    __syncthreads();
  }
}

// ------------------------------------------------------ fused denoise stage
// One workgroup (4 wave32s) per group g:
//   gather Y (and X for stage2) -> Q = M M^T (WMMA f32 16x16x4)
//   -> (+ D*I for stage2) -> in-place Gauss-Jordan inverse (LDS)
//   -> theta = I - D*inv(Q) (symmetric) -> weights
//   -> Xhat = theta @ Y (WMMA) -> fused fold via global f32 atomics.
template <int K, int KP, bool S2>
__global__ __launch_bounds__(BS) void denoise_kernel(
    const float* __restrict__ ysrc, const float* __restrict__ xsrc,
    const int* __restrict__ ind, const float* __restrict__ sigma,
    float* __restrict__ accN, float* __restrict__ accD) {
  constexpr int LDY = 65;  // odd stride: conflict-free strided LDS access
  constexpr int LDQ = 65;
  constexpr int NT = KP / 16;
  __shared__ float Ys[KP * LDY];
  __shared__ float Xs[S2 ? KP * LDY : 1];
  __shared__ float Qs[KP * LDQ];
  __shared__ float colbuf[KP];
  __shared__ float wgt[KP];
  __shared__ int inds[K];

  const int tid = threadIdx.x;
  const int g = blockIdx.x;
  const int img = g >> 12;  // 4096 groups per image
  const size_t imgOff = (size_t)img * IMG_HW;
  const float sg = sigma[0];
  const float Dv = 49.f * sg * sg;

  for (int j = tid; j < K; j += BS) inds[j] = ind[(size_t)g * K + j];
  __syncthreads();

  // Gather patch groups into LDS (zero-padded to KP rows x 64 cols).
  for (int i = tid; i < KP * 64; i += BS) {
    int row = i >> 6, col = i & 63;
    float vy = 0.f, vx = 0.f;
    if (row < K && col < 49) {
      int pid = inds[row];
      int pr = pid / WC, pc = pid % WC;
      size_t off = imgOff + (size_t)(pr + col / 7) * IMG_W + (pc + col % 7);
      vy = ysrc[off];
      if constexpr (S2) vx = xsrc[off];
    }
    Ys[row * LDY + col] = vy;
    if constexpr (S2) Xs[row * LDY + col] = vx;
  }
  __syncthreads();

  const float* Msrc = S2 ? Xs : Ys;
  const int lane = tid & 31;
  const int wave = tid >> 5;
  const int li = lane & 15;
  const int h2 = (lane >> 4) << 1;           // A/B frag: K offset 0 or 2
  const int rsh = (lane >> 4) << 3;          // C/D frag: M offset 0 or 8

  // Q = M M^T, tiled 16x16, K = 49 padded to 52 (4 per WMMA step).
  for (int t = wave; t < NT * NT; t += 4) {
    int ti = t / NT, tj = t % NT;
    v8f acc = {};
    for (int kk = 0; kk < 52; kk += 4) {
      int kb = kk + h2;
      v2f a, b;
      a.x = Msrc[(ti * 16 + li) * LDY + kb];
      a.y = Msrc[(ti * 16 + li) * LDY + kb + 1];
      b.x = Msrc[(tj * 16 + li) * LDY + kb];
      b.y = Msrc[(tj * 16 + li) * LDY + kb + 1];
      acc = wmma4(a, b, acc);
    }
#pragma unroll
    for (int r = 0; r < 8; ++r)
      Qs[(ti * 16 + rsh + r) * LDQ + tj * 16 + li] = acc[r];
  }
  __syncthreads();

  if constexpr (S2) {
    for (int i = tid; i < K; i += BS) Qs[i * LDQ + i] += Dv;
    __syncthreads();
  }

  // In-place Gauss-Jordan inversion of the KxK block (SPD-ish, no pivoting).
  for (int pv = 0; pv < K; ++pv) {
    float dinv = 1.f / Qs[pv * LDQ + pv];
    for (int r = tid; r < K; r += BS) colbuf[r] = Qs[r * LDQ + pv];
    __syncthreads();
    for (int c = tid; c < K; c += BS)
      Qs[pv * LDQ + c] = (c == pv) ? dinv : Qs[pv * LDQ + c] * dinv;
    __syncthreads();
    for (int i = tid; i < K * K; i += BS) {
      int r = i / K, c = i % K;
      if (r != pv) {
        float f = colbuf[r];
        Qs[r * LDQ + c] =
            (c == pv) ? (-f * dinv) : (Qs[r * LDQ + c] - f * Qs[pv * LDQ + c]);
      }
    }
    __syncthreads();
  }

  // theta = I - D*inv(Q), zero-padded to KP x KP (elementwise, in place).
  for (int i = tid; i < KP * KP; i += BS) {
    int r = i / KP, c = i % KP;
    float v = 0.f;
    if (r < K && c < K)
      v = ((r == c) ? 1.f : 0.f) - Dv * Qs[r * LDQ + c];
    Qs[r * LDQ + c] = v;
  }
  __syncthreads();

  // weights[i] = 1 / sum_j theta[i][j]^2
  for (int r = tid; r < KP; r += BS) {
    float s = 0.f;
    for (int c = 0; c < K; ++c) {
      float t = Qs[r * LDQ + c];
      s += t * t;
    }
    wgt[r] = (r < K) ? 1.f / s : 0.f;
  }
  __syncthreads();

  // Xhat = theta @ Y (KP x KP @ KP x 64), fused with weighted fold-aggregate.
  for (int t = wave; t < NT * 4; t += 4) {
    int ti = t >> 2, tj = t & 3;
    v8f acc = {};
    for (int kk = 0; kk < KP; kk += 4) {
      int kb = kk + h2;
      v2f a, b;
      a.x = Qs[(ti * 16 + li) * LDQ + kb];
      a.y = Qs[(ti * 16 + li) * LDQ + kb + 1];
      b.x = Ys[kb * LDY + tj * 16 + li];
      b.y = Ys[(kb + 1) * LDY + tj * 16 + li];
      acc = wmma4(a, b, acc);
    }
    int e = tj * 16 + li;  // patch element 0..48
    if (e < 49) {
      int ea = e / 7, eb = e % 7;
#pragma unroll
      for (int r = 0; r < 8; ++r) {
        int row = ti * 16 + rsh + r;
        if (row < K) {
          float wg = wgt[row];
          int pid = inds[row];
          size_t off =
              imgOff + (size_t)(pid / WC + ea) * IMG_W + (pid % WC + eb);
          atomicAdd(&accN[off], acc[r] * wg);
          atomicAdd(&accD[off], wg);
        }
      }
    }
  }
}

__global__ void div_kernel(const float* __restrict__ num,
                           const float* __restrict__ den,
                           float* __restrict__ out, int n) {
  int i = blockIdx.x * blockDim.x + threadIdx.x;
  if (i < n) out[i] = num[i] / den[i];
}

// ------------------------------------------------------------------- driver
extern "C" void kernel_launch(void* const* d_in, const int* in_sizes, int n_in,
                              void* d_out, int out_size, void* d_ws,
                              size_t ws_size, hipStream_t stream) {
  (void)in_sizes; (void)n_in; (void)out_size; (void)ws_size;
  const float* y = (const float*)d_in[0];
  const float* sigma = (const float*)d_in[1];
  float* ws = (float*)d_ws;
  size_t o = 0;
  float* padB = ws + o; o += (size_t)NIMG * PADW * PADW;   // 341,056
  float* nrmB = ws + o; o += (size_t)NIMG * NRMW * NRMW;   // 327,184
  float* den1 = ws + o; o += (size_t)NIMG * IMG_HW;        // 262,144
  float* accN = ws + o; o += (size_t)NIMG * IMG_HW;        // contiguous with
  float* accD = ws + o; o += (size_t)NIMG * IMG_HW;        // accN (single fill)
  int* indB = (int*)(ws + o);                              // 16384*55 ints
  float* out = (float*)d_out;

  const int TPB = 256;
  const int nPad = NIMG * PADW * PADW;
  const int nNrm = NIMG * NRMW * NRMW;
  const int nImg = NIMG * IMG_HW;
  dim3 bmGrid(NPOS, NPOS, NIMG);

  // ---- stage 1 (k = 18)
  pad_kernel<<<(nPad + TPB - 1) / TPB, TPB, 0, stream>>>(y, padB);
  norm_kernel<<<(nNrm + TPB - 1) / TPB, TPB, 0, stream>>>(padB, nrmB);
  bm_kernel<18><<<bmGrid, BS, 0, stream>>>(padB, nrmB, indB);
  fill_kernel<<<(2 * nImg + TPB - 1) / TPB, TPB, 0, stream>>>(accN, 0.f, 2 * nImg);
  denoise_kernel<18, 32, false><<<16384, BS, 0, stream>>>(y, nullptr, indB,
                                                          sigma, accN, accD);
  div_kernel<<<(nImg + TPB - 1) / TPB, TPB, 0, stream>>>(accN, accD, den1, nImg);

  // ---- stage 2 (k = 55)
  pad_kernel<<<(nPad + TPB - 1) / TPB, TPB, 0, stream>>>(den1, padB);
  norm_kernel<<<(nNrm + TPB - 1) / TPB, TPB, 0, stream>>>(padB, nrmB);
  bm_kernel<55><<<bmGrid, BS, 0, stream>>>(padB, nrmB, indB);
  fill_kernel<<<(2 * nImg + TPB - 1) / TPB, TPB, 0, stream>>>(accN, 0.f, 2 * nImg);
  denoise_kernel<55, 64, true><<<16384, BS, 0, stream>>>(y, den1, indB, sigma,
                                                         accN, accD);
  div_kernel<<<(nImg + TPB - 1) / TPB, TPB, 0, stream>>>(accN, accD, out, nImg);
}